// RecyclingEmbedder_43946105373392
// MI455X (gfx1250) — compile-verified
//
#include <hip/hip_runtime.h>
#include <hip/hip_bf16.h>

#define N_SEQ_    128
#define N_RES_    768
#define MSA_DIM_  256
#define PAIR_DIM_ 128
#define N_BINS_   15
#define EPS_      1e-5f

// ---------------------------------------------------------------------------
// wave32 butterfly sum reduction (gfx1250 is wave32-only)
// ---------------------------------------------------------------------------
__device__ __forceinline__ float wave_reduce_sum(float v) {
#pragma unroll
    for (int m = 16; m >= 1; m >>= 1)
        v += __shfl_xor(v, m, 32);
    return v;
}

// ---------------------------------------------------------------------------
// gfx1250 async global->LDS copy (ASYNCcnt path, CDNA5 ISA)
//   each lane copies 4 bytes from its global address in vaddr to its LDS address in vdst  (per-lane LDS destination)
// ---------------------------------------------------------------------------
__device__ __forceinline__ void async_g2l_b32(const void* gsrc, void* lds_dst) {
    // generic pointer to __shared__ is {aperture_hi, lds_offset}: low 32 bits
    // are the wave-relative LDS byte offset the instruction wants in VDST.
    uint32_t loff = (uint32_t)(uintptr_t)lds_dst;
    asm volatile("global_load_async_to_lds_b32 %0, %1, off"
                 :
                 : "v"(loff), "v"(gsrc)
                 : "memory");
}

__device__ __forceinline__ void async_wait_all() {
    asm volatile("s_wait_asynccnt 0x0" ::: "memory");
}

// ---------------------------------------------------------------------------
// Bulk copy of msa rows 1..127: 4 independent b128 loads in flight per wave
// before the first wait, grid-stride batches.
// ---------------------------------------------------------------------------
__global__ __launch_bounds__(256) void msa_copy_kernel(const float4* __restrict__ src,
                                                       float4* __restrict__ dst,
                                                       int n4) {
    const int tid    = blockIdx.x * blockDim.x + threadIdx.x;
    const int stride = gridDim.x * blockDim.x;   // threads
    const int batch  = stride * 4;

    for (int base = tid; base < n4; base += batch) {
        const int i0 = base;
        const int i1 = base + stride;
        const int i2 = base + 2 * stride;
        const int i3 = base + 3 * stride;
        // keep the stream ahead (speculative, safely dropped on OOB)
        __builtin_prefetch((const void*)(src + i0 + batch), 0, 0);

        float4 x0, x1, x2, x3;
        x0 = src[i0];
        if (i1 < n4) x1 = src[i1];
        if (i2 < n4) x2 = src[i2];
        if (i3 < n4) x3 = src[i3];

        dst[i0] = x0;
        if (i1 < n4) dst[i1] = x1;
        if (i2 < n4) dst[i2] = x2;
        if (i3 < n4) dst[i3] = x3;
    }
}

// ---------------------------------------------------------------------------
// msa_out[0, r, :] = x + LN(x)*g + b  for the 768 rows of sequence 0.
// One wave per 256-channel row: lane holds 2x float4.
// ---------------------------------------------------------------------------
__global__ __launch_bounds__(256) void msa_ln_kernel(const float* __restrict__ msa,
                                                     const float* __restrict__ g,
                                                     const float* __restrict__ b,
                                                     float* __restrict__ out) {
    const int wave = (int)((blockIdx.x * blockDim.x + threadIdx.x) >> 5);
    const int lane = threadIdx.x & 31;
    if (wave >= N_RES_) return;

    const float4* row = (const float4*)(msa + (size_t)wave * MSA_DIM_);
    float4 x0 = row[lane];
    float4 x1 = row[lane + 32];
    float4 g0 = ((const float4*)g)[lane];
    float4 g1 = ((const float4*)g)[lane + 32];
    float4 b0 = ((const float4*)b)[lane];
    float4 b1 = ((const float4*)b)[lane + 32];

    float s = x0.x + x0.y + x0.z + x0.w + x1.x + x1.y + x1.z + x1.w;
    s = wave_reduce_sum(s);
    const float mean = s * (1.0f / MSA_DIM_);

    float a0 = x0.x - mean, a1 = x0.y - mean, a2 = x0.z - mean, a3 = x0.w - mean;
    float a4 = x1.x - mean, a5 = x1.y - mean, a6 = x1.z - mean, a7 = x1.w - mean;
    float v = a0 * a0 + a1 * a1 + a2 * a2 + a3 * a3 +
              a4 * a4 + a5 * a5 + a6 * a6 + a7 * a7;
    v = wave_reduce_sum(v);
    const float rs = rsqrtf(v * (1.0f / MSA_DIM_) + EPS_);

    float4 o0, o1;
    o0.x = x0.x + (a0 * rs * g0.x + b0.x);
    o0.y = x0.y + (a1 * rs * g0.y + b0.y);
    o0.z = x0.z + (a2 * rs * g0.z + b0.z);
    o0.w = x0.w + (a3 * rs * g0.w + b0.w);
    o1.x = x1.x + (a4 * rs * g1.x + b1.x);
    o1.y = x1.y + (a5 * rs * g1.y + b1.y);
    o1.z = x1.z + (a6 * rs * g1.z + b1.z);
    o1.w = x1.w + (a7 * rs * g1.w + b1.w);

    float4* orow = (float4*)(out + (size_t)wave * MSA_DIM_);
    orow[lane]      = o0;
    orow[lane + 32] = o1;
}

// ---------------------------------------------------------------------------
// pair_out[i,j,:] = x + (Wt[bin] + bias) + LN(x)*g + bp
// One wave per (i,j): lane L holds channels 4L..4L+3 as a float4.
// - W transposed into LDS via async global->LDS scatter (per-lane LDS dest),
//   so the bin gather is a conflict-free ds_load_b128.
// - Register double-buffer: next row's b128 load issues before this row's
//   LN reduction chain, hiding memory latency inside the wave.
// - (i,j) wave-uniform -> Cb coordinate loads scalarize.
// ---------------------------------------------------------------------------
__global__ __launch_bounds__(256) void pair_kernel(const float* __restrict__ pair,
                                                   const float* __restrict__ cb,
                                                   const float* __restrict__ W,
                                                   const float* __restrict__ bias,
                                                   const float* __restrict__ g,
                                                   const float* __restrict__ bp,
                                                   const float* __restrict__ bins,
                                                   float* __restrict__ out) {
    __shared__ float sW[N_BINS_ * PAIR_DIM_];  // Wt[bin][c]
    __shared__ float sBins[N_BINS_];

    // async transpose-stage: W[c*15 + bin] is copied to sW[bin*128 + c]
    for (int t = threadIdx.x; t < N_BINS_ * PAIR_DIM_; t += blockDim.x) {
        const int c   = t / N_BINS_;
        const int bin = t - c * N_BINS_;
        async_g2l_b32(W + t, sW + bin * PAIR_DIM_ + c);
    }
    if (threadIdx.x < N_BINS_)
        async_g2l_b32(bins + threadIdx.x, sBins + threadIdx.x);
    async_wait_all();
    __syncthreads();

    const int lane   = threadIdx.x & 31;
    const int wave   = (int)((blockIdx.x * blockDim.x + threadIdx.x) >> 5);
    const int nwaves = (int)((gridDim.x * blockDim.x) >> 5);

    // per-lane channel parameters, loaded once, live in VGPRs for the loop
    const float4 gv  = ((const float4*)g)[lane];
    const float4 bpv = ((const float4*)bp)[lane];
    const float4 bv  = ((const float4*)bias)[lane];

    const int npairs = N_RES_ * N_RES_;
    if (wave >= npairs) return;

    // prime the pipeline
    float4 x = ((const float4*)(pair + (size_t)wave * PAIR_DIM_))[lane];

    for (int p = wave; p < npairs; p += nwaves) {
        const int pn = p + nwaves;

        // issue next row's load before this row's reduction chain
        float4 xn;
        if (pn < npairs) {
            xn = ((const float4*)(pair + (size_t)pn * PAIR_DIM_))[lane];
            __builtin_prefetch(
                (const void*)((const char*)(pair + (size_t)(pn + nwaves) * PAIR_DIM_) +
                              (size_t)lane * 16),
                0, 0);
        }

        const int i = p / N_RES_;
        const int j = p - i * N_RES_;

        // wave-uniform addresses -> scalar loads (cb is 9 KB, cache-hot)
        const float dx = cb[3 * i]     - cb[3 * j];
        const float dy = cb[3 * i + 1] - cb[3 * j + 1];
        const float dz = cb[3 * i + 2] - cb[3 * j + 2];
        const float dist = sqrtf(dx * dx + dy * dy + dz * dz);

        int   idx  = 0;
        float best = fabsf(dist - sBins[0]);
#pragma unroll
        for (int k = 1; k < N_BINS_; ++k) {
            const float a = fabsf(dist - sBins[k]);
            if (a < best) { best = a; idx = k; }
        }

        float s = x.x + x.y + x.z + x.w;
        s = wave_reduce_sum(s);
        const float mean = s * (1.0f / PAIR_DIM_);

        const float ax = x.x - mean, ay = x.y - mean, az = x.z - mean, aw = x.w - mean;
        float v = ax * ax + ay * ay + az * az + aw * aw;
        v = wave_reduce_sum(v);
        const float rs = rsqrtf(v * (1.0f / PAIR_DIM_) + EPS_);

        const float4 wv = ((const float4*)(sW + idx * PAIR_DIM_))[lane];

        float4 o;
        o.x = x.x + wv.x + bv.x + (ax * rs * gv.x + bpv.x);
        o.y = x.y + wv.y + bv.y + (ay * rs * gv.y + bpv.y);
        o.z = x.z + wv.z + bv.z + (az * rs * gv.z + bpv.z);
        o.w = x.w + wv.w + bv.w + (aw * rs * gv.w + bpv.w);

        ((float4*)(out + (size_t)p * PAIR_DIM_))[lane] = o;

        x = xn;
    }
}

// ---------------------------------------------------------------------------
extern "C" void kernel_launch(void* const* d_in, const int* in_sizes, int n_in,
                              void* d_out, int out_size, void* d_ws, size_t ws_size,
                              hipStream_t stream) {
    (void)in_sizes; (void)n_in; (void)out_size; (void)d_ws; (void)ws_size;

    const float* msa    = (const float*)d_in[0];  // [128,768,256]
    const float* pair   = (const float*)d_in[1];  // [768,768,128]
    const float* cb     = (const float*)d_in[2];  // [768,3]
    const float* W      = (const float*)d_in[3];  // [128,15]
    const float* b      = (const float*)d_in[4];  // [128]
    const float* g_pair = (const float*)d_in[5];  // [128]
    const float* b_pair = (const float*)d_in[6];  // [128]
    const float* g_msa  = (const float*)d_in[7];  // [256]
    const float* b_msa  = (const float*)d_in[8];  // [256]
    const float* bins   = (const float*)d_in[9];  // [15]

    float* out      = (float*)d_out;
    float* msa_out  = out;                                            // 128*768*256
    float* pair_out = out + (size_t)N_SEQ_ * N_RES_ * MSA_DIM_;       // 768*768*128

    // msa rows 1..127: straight copy (float4, 4-deep batched grid-stride)
    {
        const size_t row_elems = (size_t)N_RES_ * MSA_DIM_;
        const int n4 = (int)(((size_t)(N_SEQ_ - 1) * row_elems) / 4);
        msa_copy_kernel<<<6144, 256, 0, stream>>>(
            (const float4*)(msa + row_elems), (float4*)(msa_out + row_elems), n4);
    }

    // msa row 0: x + LN(x); 768 rows, one wave each -> 96 blocks of 8 waves
    msa_ln_kernel<<<96, 256, 0, stream>>>(msa, g_msa, b_msa, msa_out);

    // pair: 589824 rows, 18432 waves (2304 blocks x 8 waves) -> 32 rows/wave
    pair_kernel<<<2304, 256, 0, stream>>>(pair, cb, W, b, g_pair, b_pair, bins, pair_out);
}